// GraphEncoder_71107478552909
// MI455X (gfx1250) — compile-verified
//
#include <hip/hip_runtime.h>
#include <hip/hip_bf16.h>

#define D_IN   128
#define D_HID  64
#define D_OUT  64
#define NEG_SLOPE 0.2f

typedef float v2f __attribute__((ext_vector_type(2)));
typedef float v8f __attribute__((ext_vector_type(8)));

// ---------------------------------------------------------------------------
// GEMM: C[N x 64] = A[N x K] * B[K x 64] using V_WMMA_F32_16X16X4_F32.
// One wave computes a 16x16 tile. Block = 256 threads = 8 waves = 32 rows.
// A-frag (16x4 f32): lanes 0-15 hold M=0..15 {K=k0,k0+1}; lanes 16-31 {K=k0+2,k0+3}.
// B-frag (4x16 f32): mirrored layout over N.
// D (16x16 f32, 8 VGPRs): VGPR j -> M=j (lanes 0-15) / M=j+8 (lanes 16-31), N=lane&15.
// ---------------------------------------------------------------------------
template <int K>
__global__ void gemm_wmma_f32(const float* __restrict__ A,
                              const float* __restrict__ B,
                              float* __restrict__ C, int nrows) {
  const int wid  = threadIdx.x >> 5;
  const int lane = threadIdx.x & 31;
  const int rowTile = blockIdx.x * 2 + (wid >> 2);   // 16-row tile index
  const int colTile = wid & 3;                       // 4 tiles cover 64 cols
  const int m    = lane & 15;
  const int half = lane >> 4;

  int row = rowTile * 16 + m;
  int rowClamped = row < nrows ? row : (nrows - 1);
  const float* arow = A + (size_t)rowClamped * K;
  const int ncol = colTile * 16 + m;

  v8f acc = {0.f, 0.f, 0.f, 0.f, 0.f, 0.f, 0.f, 0.f};
#pragma unroll
  for (int k0 = 0; k0 < K; k0 += 4) {
    const int ka = k0 + 2 * half;
    v2f a, b;
    a.x = arow[ka];
    a.y = arow[ka + 1];
    b.x = B[ka * 64 + ncol];
    b.y = B[(ka + 1) * 64 + ncol];
    acc = __builtin_amdgcn_wmma_f32_16x16x4_f32(
        false, a, false, b, (short)0, acc, false, false);
  }
#pragma unroll
  for (int j = 0; j < 8; ++j) {
    int rrow = rowTile * 16 + j + 8 * half;
    if (rrow < nrows) C[(size_t)rrow * 64 + ncol] = acc[j];
  }
}

// ---------------------------------------------------------------------------
// Per-node attention dots: asrc[n] = h[n,:]·att_s ; adst[n] = h[n,:]·att_d
// One wave per node, 2 features per lane, shfl-tree reduce (wave32).
// ---------------------------------------------------------------------------
__global__ void node_dots(const float* __restrict__ h,
                          const float* __restrict__ att_s,
                          const float* __restrict__ att_d,
                          float* __restrict__ asrc, float* __restrict__ adst,
                          int N) {
  int n    = (blockIdx.x * blockDim.x + threadIdx.x) >> 5;
  int lane = threadIdx.x & 31;
  if (n >= N) return;
  const float* row = h + (size_t)n * 64;
  float h0 = row[lane], h1 = row[lane + 32];
  float ps = h0 * att_s[lane] + h1 * att_s[lane + 32];
  float pd = h0 * att_d[lane] + h1 * att_d[lane + 32];
#pragma unroll
  for (int off = 16; off > 0; off >>= 1) {
    ps += __shfl_down(ps, off, 32);
    pd += __shfl_down(pd, off, 32);
  }
  if (lane == 0) {
    asrc[n] = ps;
    adst[n] = pd;
  }
}

__device__ __forceinline__ unsigned int f32_orderkey(float f) {
  unsigned int b = __float_as_uint(f);
  return (b & 0x80000000u) ? ~b : (b | 0x80000000u);
}
__device__ __forceinline__ float f32_unorderkey(unsigned int k) {
  unsigned int b = (k & 0x80000000u) ? (k ^ 0x80000000u) : ~k;
  return __uint_as_float(b);
}

// Pass A: e = leaky_relu(asrc[s] + adst[d]); store e; atomic-max key into emax[d].
__global__ void edge_logits(const int* __restrict__ src, const int* __restrict__ dst,
                            const float* __restrict__ asrc, const float* __restrict__ adst,
                            float* __restrict__ e_edge, unsigned int* __restrict__ emax,
                            int E, int N) {
  int i = blockIdx.x * blockDim.x + threadIdx.x;
  int EA = E + N;
  if (i >= EA) return;
  int s = (i < E) ? src[i] : (i - E);
  int d = (i < E) ? dst[i] : (i - E);
  float e = asrc[s] + adst[d];
  e = (e > 0.f) ? e : NEG_SLOPE * e;
  e_edge[i] = e;
  atomicMax(&emax[d], f32_orderkey(e));
}

// Pass B: w = exp(e - emax[d]); store w over e; denom[d] += w.
__global__ void edge_weights(const int* __restrict__ src, const int* __restrict__ dst,
                             float* __restrict__ e_edge,
                             const unsigned int* __restrict__ emax,
                             float* __restrict__ denom, int E, int N) {
  int i = blockIdx.x * blockDim.x + threadIdx.x;
  int EA = E + N;
  if (i >= EA) return;
  int d = (i < E) ? dst[i] : (i - E);
  float m = f32_unorderkey(emax[d]);
  float w = __expf(e_edge[i] - m);
  e_edge[i] = w;
  atomicAdd(&denom[d], w);
}

// Pass C: warp per edge; alpha = w/denom[d]; out[d,:] += alpha*h[s,:] (2 feats/lane).
__global__ void edge_aggregate(const int* __restrict__ src, const int* __restrict__ dst,
                               const float* __restrict__ w_edge,
                               const float* __restrict__ denom,
                               const float* __restrict__ h,
                               float* __restrict__ out, int E, int N) {
  int gw   = (blockIdx.x * blockDim.x + threadIdx.x) >> 5;
  int lane = threadIdx.x & 31;
  int EA = E + N;
  if (gw >= EA) return;
  int s = (gw < E) ? src[gw] : (gw - E);
  int d = (gw < E) ? dst[gw] : (gw - E);
  float alpha = w_edge[gw] / (denom[d] + 1e-16f);
  const float* hs = h + (size_t)s * 64;
  float* od = out + (size_t)d * 64;
  atomicAdd(&od[lane],      alpha * hs[lane]);
  atomicAdd(&od[lane + 32], alpha * hs[lane + 32]);
}

// h2 = relu(out1 + b1)
__global__ void bias_relu(const float* __restrict__ in, const float* __restrict__ b,
                          float* __restrict__ out, int total) {
  int i = blockIdx.x * blockDim.x + threadIdx.x;
  if (i >= total) return;
  float v = in[i] + b[i & 63];
  out[i] = v > 0.f ? v : 0.f;
}

// d_out[f] = b2[f]   (mean of (out2+b2) = mean(out2)+b2)
__global__ void init_out(const float* __restrict__ b2, float* __restrict__ dout) {
  int t = threadIdx.x;
  if (t < 64) dout[t] = b2[t];
}

// d_out[f] += sum_rows(out2[:,f]) / N  via block-local LDS partials.
__global__ void mean_accum(const float* __restrict__ out2, float* __restrict__ dout,
                           int N) {
  __shared__ float s[64];
  int tid = threadIdx.x;
  if (tid < 64) s[tid] = 0.f;
  __syncthreads();
  int f = tid & 63;
  int r0 = blockIdx.x * 4 + (tid >> 6);
  float acc = 0.f;
  for (int r = r0; r < N; r += gridDim.x * 4) acc += out2[(size_t)r * 64 + f];
  atomicAdd(&s[f], acc);
  __syncthreads();
  if (tid < 64) atomicAdd(&dout[tid], s[tid] / (float)N);
}

extern "C" void kernel_launch(void* const* d_in, const int* in_sizes, int n_in,
                              void* d_out, int out_size, void* d_ws, size_t ws_size,
                              hipStream_t stream) {
  const float* x        = (const float*)d_in[0];
  const int*   eidx     = (const int*)d_in[1];   // [2, E] int32 (JAX default x64-off)
  /* d_in[2] = edge_attr, unused by reference */
  const float* W1       = (const float*)d_in[3];
  const float* att_src1 = (const float*)d_in[4];
  const float* att_dst1 = (const float*)d_in[5];
  const float* b1       = (const float*)d_in[6];
  const float* W2       = (const float*)d_in[7];
  const float* att_src2 = (const float*)d_in[8];
  const float* att_dst2 = (const float*)d_in[9];
  const float* b2       = (const float*)d_in[10];
  float* dout = (float*)d_out;

  const int N  = in_sizes[0] / D_IN;
  const int E  = in_sizes[1] / 2;
  const int EA = E + N;
  const int* src = eidx;
  const int* dst = eidx + E;

  // ---- workspace carve ----
  float* bufA = (float*)d_ws;               // N*64 : h1, then h2, then out2
  float* bufB = bufA + (size_t)N * 64;      // N*64 : out1, then h2'
  float* asrc = bufB + (size_t)N * 64;      // N
  float* adst = asrc + N;                   // N
  float* denom = adst + N;                  // N
  unsigned int* emax = (unsigned int*)(denom + N);  // N
  float* eedge = (float*)(emax + N);        // E+N

  const int TPB = 256;
  dim3 blk(TPB);
  dim3 gemmGrid((N + 31) / 32);
  dim3 nodeWaveGrid((N + 7) / 8);           // 8 waves/block, wave per node
  dim3 edgeGrid((EA + TPB - 1) / TPB);
  dim3 edgeWarpGrid((EA + 7) / 8);          // warp per edge
  dim3 elemGrid((N * 64 + TPB - 1) / TPB);

  // =================== Layer 1 ===================
  gemm_wmma_f32<D_IN><<<gemmGrid, blk, 0, stream>>>(x, W1, bufA, N);
  node_dots<<<nodeWaveGrid, blk, 0, stream>>>(bufA, att_src1, att_dst1, asrc, adst, N);

  hipMemsetAsync(emax, 0, (size_t)N * 4, stream);      // key 0 < key(-inf)
  hipMemsetAsync(denom, 0, (size_t)N * 4, stream);
  hipMemsetAsync(bufB, 0, (size_t)N * 64 * 4, stream);

  edge_logits<<<edgeGrid, blk, 0, stream>>>(src, dst, asrc, adst, eedge, emax, E, N);
  edge_weights<<<edgeGrid, blk, 0, stream>>>(src, dst, eedge, emax, denom, E, N);
  edge_aggregate<<<edgeWarpGrid, blk, 0, stream>>>(src, dst, eedge, denom, bufA, bufB, E, N);

  bias_relu<<<elemGrid, blk, 0, stream>>>(bufB, b1, bufA, N * 64);   // bufA = h2

  // =================== Layer 2 ===================
  gemm_wmma_f32<D_HID><<<gemmGrid, blk, 0, stream>>>(bufA, W2, bufB, N);  // bufB = h2'
  node_dots<<<nodeWaveGrid, blk, 0, stream>>>(bufB, att_src2, att_dst2, asrc, adst, N);

  hipMemsetAsync(emax, 0, (size_t)N * 4, stream);
  hipMemsetAsync(denom, 0, (size_t)N * 4, stream);
  hipMemsetAsync(bufA, 0, (size_t)N * 64 * 4, stream);  // bufA = out2

  edge_logits<<<edgeGrid, blk, 0, stream>>>(src, dst, asrc, adst, eedge, emax, E, N);
  edge_weights<<<edgeGrid, blk, 0, stream>>>(src, dst, eedge, emax, denom, E, N);
  edge_aggregate<<<edgeWarpGrid, blk, 0, stream>>>(src, dst, eedge, denom, bufB, bufA, E, N);

  // =================== Mean pool (+b2) ===================
  init_out<<<1, 64, 0, stream>>>(b2, dout);
  mean_accum<<<1024, blk, 0, stream>>>(bufA, dout, N);
}